// CharRNN_62225486184551
// MI455X (gfx1250) — compile-verified
//
#include <hip/hip_runtime.h>
#include <hip/hip_bf16.h>

#define T_LEN 512
#define B_SZ  512
#define EMB   50
#define HID   30
#define G4    120
#define VOC   100
#define NP    128   // padded gate width (8 x 16-col WMMA tiles)
#define KE    52    // padded embed K (mult of 4)
#define KH    32    // padded hidden K (mult of 4)
#define VP    112   // padded vocab width (7 tiles)
#define CHUNK 16    // batch rows per workgroup (one WMMA M-tile)
#define NT    128   // 4 wave32s

#define NP2 (NP * 2)
#define KH2 (KH * 2)
#define VP2 (VP * 2)

typedef __attribute__((ext_vector_type(2))) float v2f;
typedef __attribute__((ext_vector_type(8))) float v8f;

__device__ __forceinline__ v8f wmma4(v2f a, v2f b, v8f c) {
  // V_WMMA_F32_16X16X4_F32: D = A(16x4) * B(4x16) + C(16x16), fp32 tensor op
  return __builtin_amdgcn_wmma_f32_16x16x4_f32(false, a, false, b, (short)0, c,
                                               false, false);
}

__device__ __forceinline__ float sigm(float x) { return 1.0f / (1.0f + __expf(-x)); }

// acc += A[16 x kdim] * B[kdim x N] for one 16-col tile.
// aPtr: per-lane A row, already offset by khalf (lanes 0-15 read K=kb..kb+1,
//       lanes 16-31 read K=kb+2..kb+3) -> single ds_load_b64 each.
// bPtr: per-lane column pointer into K-pair-interleaved B ([K/2][N][2]),
//       already offset by n*2 + (khalf>>1)*ld2 -> single ds_load_b64 each,
//       immediate offsets after unroll.
__device__ __forceinline__ v8f mm_acc(v8f acc, const float* __restrict__ aPtr, int kdim,
                                      const float* __restrict__ bPtr, int ld2) {
  #pragma unroll
  for (int kb = 0; kb < kdim; kb += 4) {
    v2f a = *(const v2f*)(aPtr + kb);
    v2f b = *(const v2f*)(bPtr + (kb >> 1) * ld2);
    acc = wmma4(a, b, acc);
  }
  return acc;
}

extern "C" __global__ __launch_bounds__(NT)
void charnn_fused(const int* __restrict__ batch_x, const int* __restrict__ lens,
                  const float* __restrict__ emb,
                  const float* __restrict__ w_ih1, const float* __restrict__ w_hh1,
                  const float* __restrict__ b_ih1, const float* __restrict__ b_hh1,
                  const float* __restrict__ w_ih2, const float* __restrict__ w_hh2,
                  const float* __restrict__ b_ih2, const float* __restrict__ b_hh2,
                  const float* __restrict__ lw1,  const float* __restrict__ lb1,
                  const float* __restrict__ lw2,  const float* __restrict__ lb2,
                  float* __restrict__ out)
{
  extern __shared__ float smem[];
  const int tid   = threadIdx.x;
  const int lane  = tid & 31;
  const int wave  = tid >> 5;
  const int row   = lane & 15;
  const int khalf = (lane < 16) ? 0 : 2;  // K sub-block per lane half
  const int kh2   = khalf >> 1;           // pair index offset
  const int mhalf = (lane < 16) ? 0 : 8;  // C/D row split at lane 16
  const int b0    = blockIdx.x * CHUNK;

  // ---- LDS carve-up (all weights resident for the whole recurrence) ----
  // B matrices stored K-pair interleaved: P[(k>>1)*ld2 + n*2 + (k&1)] = W^T[k][n]
  float* embL = smem;                  // [VOC][KE] row-major
  float* wi1  = embL + VOC * KE;       // [KE/2][NP][2]
  float* wh1  = wi1  + KE * NP;        // [KH/2][NP][2]
  float* wi2  = wh1  + KH * NP;        // [KH/2][NP][2]
  float* wh2  = wi2  + KH * NP;        // [KH/2][NP][2]
  float* l1T  = wh2  + KH * NP;        // [KH/2][KH][2]
  float* l2T  = l1T  + KH * KH;        // [KH/2][VP][2]
  float* bg1  = l2T  + KH * VP;        // [NP] combined gate bias L1
  float* bg2  = bg1  + NP;             // [NP] combined gate bias L2
  float* bl1L = bg2  + NP;             // [KH]
  float* bl2L = bl1L + KH;             // [VP]
  float* h1   = bl2L + VP;             // [CHUNK][KH]
  float* h2   = h1   + CHUNK * KH;     // [CHUNK][KH]
  float* h2m  = h2   + CHUNK * KH;     // [CHUNK][KH] masked h2 for head
  float* hmid = h2m  + CHUNK * KH;     // [CHUNK][KH]
  float* gbuf = hmid + CHUNK * KH;     // [CHUNK][NP] gate scratch
  float* mskf = gbuf + CHUNK * NP;     // [CHUNK]
  int*   ids   = (int*)(mskf + CHUNK); // [CHUNK]
  int*   lensL = ids + CHUNK;          // [CHUNK]

  // ---- stage weights into LDS (transposed, K-pair interleaved, zero-padded) ----
  for (int i = tid; i < VOC * KE; i += NT) { int r = i / KE, k = i - r * KE;
    embL[i] = (k < EMB) ? emb[r * EMB + k] : 0.f; }
  for (int i = tid; i < KE * NP; i += NT) { int k = i / NP, n = i - k * NP;
    wi1[(k >> 1) * NP2 + n * 2 + (k & 1)] =
        (k < EMB && n < G4) ? w_ih1[n * EMB + k] : 0.f; }
  for (int i = tid; i < KH * NP; i += NT) { int k = i / NP, n = i - k * NP;
    const bool ok = (k < HID && n < G4);
    const int  d  = (k >> 1) * NP2 + n * 2 + (k & 1);
    wh1[d] = ok ? w_hh1[n * HID + k] : 0.f;
    wi2[d] = ok ? w_ih2[n * HID + k] : 0.f;
    wh2[d] = ok ? w_hh2[n * HID + k] : 0.f; }
  for (int i = tid; i < KH * KH; i += NT) { int k = i / KH, j = i - k * KH;
    l1T[(k >> 1) * KH2 + j * 2 + (k & 1)] =
        (k < HID && j < HID) ? lw1[j * HID + k] : 0.f; }
  for (int i = tid; i < KH * VP; i += NT) { int k = i / VP, v = i - k * VP;
    l2T[(k >> 1) * VP2 + v * 2 + (k & 1)] =
        (k < HID && v < VOC) ? lw2[v * HID + k] : 0.f; }
  for (int i = tid; i < NP; i += NT) {
    bg1[i] = (i < G4) ? (b_ih1[i] + b_hh1[i]) : 0.f;
    bg2[i] = (i < G4) ? (b_ih2[i] + b_hh2[i]) : 0.f; }
  for (int i = tid; i < KH; i += NT) bl1L[i] = (i < HID) ? lb1[i] : 0.f;
  for (int i = tid; i < VP; i += NT) bl2L[i] = (i < VOC) ? lb2[i] : 0.f;
  for (int i = tid; i < CHUNK * KH; i += NT) { h1[i] = 0.f; h2[i] = 0.f; h2m[i] = 0.f; hmid[i] = 0.f; }
  if (tid < CHUNK) lensL[tid] = lens[b0 + tid];

  // per-thread cell state: thread owns up to 4 (m,j) cells, fixed for all t
  float c1a[4] = {0.f, 0.f, 0.f, 0.f};
  float c2a[4] = {0.f, 0.f, 0.f, 0.f};

  __syncthreads();

  for (int t = 0; t < T_LEN; ++t) {
    if (tid < CHUNK) {
      ids[tid]  = batch_x[(b0 + tid) * T_LEN + t];
      mskf[tid] = (t < lensL[tid]) ? 1.f : 0.f;
    }
    __syncthreads();

    // ---- gates1 = bg1 + emb[ids] @ w_ih1^T + h1 @ w_hh1^T  (8 N-tiles / 4 waves) ----
    {
      const float* arowE = embL + ids[row] * KE + khalf;
      const float* arowH = h1 + row * KH + khalf;
      for (int nt = wave; nt < NP / 16; nt += 4) {
        const int n = nt * 16 + row;
        const float bv = bg1[n];
        v8f acc;
        #pragma unroll
        for (int r = 0; r < 8; ++r) acc[r] = bv;
        acc = mm_acc(acc, arowE, KE, wi1 + n * 2 + kh2 * NP2, NP2);
        acc = mm_acc(acc, arowH, KH, wh1 + n * 2 + kh2 * NP2, NP2);
        #pragma unroll
        for (int r = 0; r < 8; ++r) gbuf[(r + mhalf) * NP + n] = acc[r];
      }
    }
    __syncthreads();

    // ---- LSTM1 elementwise: i,f,g,o -> c,h with packed-sequence mask freeze ----
    #pragma unroll
    for (int r = 0; r < 4; ++r) {
      const int p = tid + NT * r;
      if (p < CHUNK * HID) {
        const int m = p / HID, j = p - m * HID;
        const float ig = sigm(gbuf[m * NP + j]);
        const float fg = sigm(gbuf[m * NP + HID + j]);
        const float gg = tanhf(gbuf[m * NP + 2 * HID + j]);
        const float og = sigm(gbuf[m * NP + 3 * HID + j]);
        const float cn = fg * c1a[r] + ig * gg;
        const float hn = og * tanhf(cn);
        const float mk = mskf[m];
        c1a[r] = mk * cn + (1.f - mk) * c1a[r];
        h1[m * KH + j] = mk * hn + (1.f - mk) * h1[m * KH + j];
      }
    }
    __syncthreads();

    // ---- gates2 = bg2 + h1 @ w_ih2^T + h2 @ w_hh2^T ----
    {
      const float* arowA = h1 + row * KH + khalf;
      const float* arowB = h2 + row * KH + khalf;
      for (int nt = wave; nt < NP / 16; nt += 4) {
        const int n = nt * 16 + row;
        const float bv = bg2[n];
        v8f acc;
        #pragma unroll
        for (int r = 0; r < 8; ++r) acc[r] = bv;
        acc = mm_acc(acc, arowA, KH, wi2 + n * 2 + kh2 * NP2, NP2);
        acc = mm_acc(acc, arowB, KH, wh2 + n * 2 + kh2 * NP2, NP2);
        #pragma unroll
        for (int r = 0; r < 8; ++r) gbuf[(r + mhalf) * NP + n] = acc[r];
      }
    }
    __syncthreads();

    // ---- LSTM2 elementwise + masked copy for head ----
    #pragma unroll
    for (int r = 0; r < 4; ++r) {
      const int p = tid + NT * r;
      if (p < CHUNK * HID) {
        const int m = p / HID, j = p - m * HID;
        const float ig = sigm(gbuf[m * NP + j]);
        const float fg = sigm(gbuf[m * NP + HID + j]);
        const float gg = tanhf(gbuf[m * NP + 2 * HID + j]);
        const float og = sigm(gbuf[m * NP + 3 * HID + j]);
        const float cn = fg * c2a[r] + ig * gg;
        const float hn = og * tanhf(cn);
        const float mk = mskf[m];
        c2a[r] = mk * cn + (1.f - mk) * c2a[r];
        const float hv = mk * hn + (1.f - mk) * h2[m * KH + j];
        h2[m * KH + j]  = hv;
        h2m[m * KH + j] = mk * hv;   // reference: out = where(mask, h2, 0) before head
      }
    }
    __syncthreads();

    // ---- head stage 1: hmid = relu(h2m @ lw1^T + lb1)  (2 tiles: waves 0,1) ----
    if (wave < 2) {
      const float* arow = h2m + row * KH + khalf;
      const int n = wave * 16 + row;
      const float bv = bl1L[n];
      v8f acc;
      #pragma unroll
      for (int r = 0; r < 8; ++r) acc[r] = bv;
      acc = mm_acc(acc, arow, KH, l1T + n * 2 + kh2 * KH2, KH2);
      #pragma unroll
      for (int r = 0; r < 8; ++r) hmid[(r + mhalf) * KH + n] = fmaxf(acc[r], 0.f);
    }
    __syncthreads();

    // ---- head stage 2: out[b0..b0+15, t, :] = hmid @ lw2^T + lb2  (7 tiles) ----
    {
      const float* arow = hmid + row * KH + khalf;
      for (int nt = wave; nt < VP / 16; nt += 4) {
        const int n = nt * 16 + row;
        const float bv = bl2L[n];
        v8f acc;
        #pragma unroll
        for (int r = 0; r < 8; ++r) acc[r] = bv;
        acc = mm_acc(acc, arow, KH, l2T + n * 2 + kh2 * VP2, VP2);
        if (n < VOC) {
          #pragma unroll
          for (int r = 0; r < 8; ++r) {
            const int m = r + mhalf;
            out[((size_t)(b0 + m) * T_LEN + t) * VOC + n] = acc[r];
          }
        }
      }
    }
    __syncthreads();
  }
}

extern "C" void kernel_launch(void* const* d_in, const int* in_sizes, int n_in,
                              void* d_out, int out_size, void* d_ws, size_t ws_size,
                              hipStream_t stream) {
  const int*   batch_x = (const int*)  d_in[0];
  const int*   lens    = (const int*)  d_in[1];
  const float* emb     = (const float*)d_in[2];
  const float* w_ih1   = (const float*)d_in[3];
  const float* w_hh1   = (const float*)d_in[4];
  const float* b_ih1   = (const float*)d_in[5];
  const float* b_hh1   = (const float*)d_in[6];
  const float* w_ih2   = (const float*)d_in[7];
  const float* w_hh2   = (const float*)d_in[8];
  const float* b_ih2   = (const float*)d_in[9];
  const float* b_hh2   = (const float*)d_in[10];
  const float* lw1     = (const float*)d_in[11];
  const float* lb1     = (const float*)d_in[12];
  const float* lw2     = (const float*)d_in[13];
  const float* lb2     = (const float*)d_in[14];
  float* out = (float*)d_out;

  constexpr size_t SMEM_FLOATS =
      (size_t)VOC * KE + (size_t)KE * NP + 3u * KH * NP + (size_t)KH * KH +
      (size_t)KH * VP + 2u * NP + KH + VP + 4u * CHUNK * KH + (size_t)CHUNK * NP +
      3u * CHUNK;
  constexpr size_t SMEM_BYTES = SMEM_FLOATS * sizeof(float);

  // ~130 KB dynamic LDS per workgroup (CDNA5 WGP has 320 KB): opt in
  (void)hipFuncSetAttribute((const void*)charnn_fused,
                            hipFuncAttributeMaxDynamicSharedMemorySize,
                            (int)SMEM_BYTES);

  charnn_fused<<<dim3(B_SZ / CHUNK), dim3(NT), SMEM_BYTES, stream>>>(
      batch_x, lens, emb, w_ih1, w_hh1, b_ih1, b_hh1,
      w_ih2, w_hh2, b_ih2, b_hh2, lw1, lb1, lw2, lb2, out);
}